// VectorizedAdaCoFSampler_58703613002437
// MI455X (gfx1250) — compile-verified
//
#include <hip/hip_runtime.h>

#define BLOCK 128
#define K2    25
#define B_    4
#define C_    3
#define H_    512
#define W_    512
#define HW    (H_ * W_)

#if defined(__HIP_DEVICE_COMPILE__) && __has_builtin(__builtin_amdgcn_global_load_async_to_lds_b32)
#define HAS_ASYNC 1
#else
#define HAS_ASYNC 0
#endif

typedef float v4f __attribute__((ext_vector_type(4)));
struct __attribute__((packed, aligned(4))) v4u { v4f v; };

typedef __attribute__((address_space(1))) int gas_int;  // global
typedef __attribute__((address_space(3))) int las_int;  // LDS

__device__ __forceinline__ v4f load4u(const float* p) {
  // 4-byte-aligned 4xfloat load: lowers to global_load_b128 (unaligned-capable)
  return ((const v4u*)p)->v;
}

__device__ __forceinline__ void async_g2l(const float* g, float* l) {
#if HAS_ASYNC
  __builtin_amdgcn_global_load_async_to_lds_b32(
      (gas_int*)g, (las_int*)l, /*offset=*/0, /*cpol=*/0);
#else
  *l = *g;  // synchronous fallback
#endif
}

__device__ __forceinline__ void wait_async0() {
#if HAS_ASYNC
#if __has_builtin(__builtin_amdgcn_s_wait_asynccnt)
  __builtin_amdgcn_s_wait_asynccnt(0);
#else
  asm volatile("s_wait_asynccnt 0" ::: "memory");
#endif
#endif
}

__device__ __forceinline__ int iclamp(int v, int lo, int hi) {
  return v < lo ? lo : (v > hi ? hi : v);
}

__device__ __forceinline__ void cubic_w(float t, float& w0, float& w1, float& w2, float& w3) {
  const float A = -0.75f;
  float t1 = t + 1.0f;
  w0 = ((A * t1 - 5.0f * A) * t1 + 8.0f * A) * t1 - 4.0f * A;
  w1 = ((A + 2.0f) * t - (A + 3.0f)) * t * t + 1.0f;
  float u = 1.0f - t;
  w2 = ((A + 2.0f) * u - (A + 3.0f)) * u * u + 1.0f;
  w3 = 1.0f - w0 - w1 - w2;
}

__global__ __launch_bounds__(BLOCK) void adacof_kernel(
    const float* __restrict__ img,
    const float* __restrict__ offs,
    const float* __restrict__ wts,
    float* __restrict__ out)
{
  __shared__ float smem[3 * K2 * BLOCK];  // 75 planes x 128 px = 38.4 KB

  const int tid   = threadIdx.x;
  const int pixel = blockIdx.x * BLOCK + tid;
  const int b  = pixel >> 18;          // / (512*512)
  const int hw = pixel & (HW - 1);
  const int w  = hw & (W_ - 1);
  const int h  = hw >> 9;

  const int obase = b * (2 * K2) * HW + hw;   // offsets plane 0 for this pixel
  const int wbase = b * K2 * HW + hw;         // weights plane 0

  // Stage all 75 streaming planes for this pixel tile into LDS asynchronously.
#pragma unroll 1
  for (int k = 0; k < K2; ++k) {
    async_g2l(offs + obase + (2 * k)     * HW, smem + (3 * k + 0) * BLOCK + tid);
    async_g2l(offs + obase + (2 * k + 1) * HW, smem + (3 * k + 1) * BLOCK + tid);
    async_g2l(wts  + wbase + k           * HW, smem + (3 * k + 2) * BLOCK + tid);
  }

  const float* img_b = img + b * C_ * HW;
  // Warm the cache for the local image neighborhood (global_prefetch_b8).
  __builtin_prefetch(img_b + hw, 0, 1);
  __builtin_prefetch(img_b + HW + hw, 0, 1);
  __builtin_prefetch(img_b + 2 * HW + hw, 0, 1);

  wait_async0();

  float acc0 = 0.0f, acc1 = 0.0f, acc2 = 0.0f;
  const float wm1 = (float)(W_ - 1);
  const float hm1 = (float)(H_ - 1);

#pragma unroll 1
  for (int k = 0; k < K2; ++k) {
    float dx = smem[(3 * k + 0) * BLOCK + tid];
    float dy = smem[(3 * k + 1) * BLOCK + tid];
    float wt = smem[(3 * k + 2) * BLOCK + tid];

    float sx = (float)w + dx;
    float sy = (float)h + dy;
    // faithful normalize/denormalize round trip from the reference
    sx = (2.0f * sx / wm1 - 1.0f + 1.0f) * 0.5f * wm1;
    sy = (2.0f * sy / hm1 - 1.0f + 1.0f) * 0.5f * hm1;

    float fx = floorf(sx), fy = floorf(sy);
    float tx = sx - fx,    ty = sy - fy;

    float wx0, wx1, wx2, wx3, wy0, wy1, wy2, wy3;
    cubic_w(tx, wx0, wx1, wx2, wx3);
    cubic_w(ty, wy0, wy1, wy2, wy3);
    wy0 *= wt; wy1 *= wt; wy2 *= wt; wy3 *= wt;
    float wyv[4] = {wy0, wy1, wy2, wy3};

    int ix0 = (int)fx - 1;
    int iy0 = (int)fy - 1;

    if ((unsigned)ix0 <= (unsigned)(W_ - 4) && (unsigned)iy0 <= (unsigned)(H_ - 4)) {
      // Interior fast path: 4 consecutive x-taps per row -> one b128 per row/channel.
      const float* p = img_b + iy0 * W_ + ix0;
#pragma unroll
      for (int j = 0; j < 4; ++j) {
        v4f r0 = load4u(p + j * W_);
        v4f r1 = load4u(p + HW + j * W_);
        v4f r2 = load4u(p + 2 * HW + j * W_);
        float d0 = r0.x * wx0 + r0.y * wx1 + r0.z * wx2 + r0.w * wx3;
        float d1 = r1.x * wx0 + r1.y * wx1 + r1.z * wx2 + r1.w * wx3;
        float d2 = r2.x * wx0 + r2.y * wx1 + r2.z * wx2 + r2.w * wx3;
        acc0 = fmaf(wyv[j], d0, acc0);
        acc1 = fmaf(wyv[j], d1, acc1);
        acc2 = fmaf(wyv[j], d2, acc2);
      }
    } else {
      // Border path: clamped scalar taps.
      int xs0 = iclamp(ix0,     0, W_ - 1);
      int xs1 = iclamp(ix0 + 1, 0, W_ - 1);
      int xs2 = iclamp(ix0 + 2, 0, W_ - 1);
      int xs3 = iclamp(ix0 + 3, 0, W_ - 1);
#pragma unroll
      for (int j = 0; j < 4; ++j) {
        int yy = iclamp(iy0 + j, 0, H_ - 1);
        const float* row = img_b + yy * W_;
        float s0 = row[xs0] * wx0 + row[xs1] * wx1 + row[xs2] * wx2 + row[xs3] * wx3;
        float s1 = row[HW + xs0] * wx0 + row[HW + xs1] * wx1 +
                   row[HW + xs2] * wx2 + row[HW + xs3] * wx3;
        float s2 = row[2 * HW + xs0] * wx0 + row[2 * HW + xs1] * wx1 +
                   row[2 * HW + xs2] * wx2 + row[2 * HW + xs3] * wx3;
        acc0 = fmaf(wyv[j], s0, acc0);
        acc1 = fmaf(wyv[j], s1, acc1);
        acc2 = fmaf(wyv[j], s2, acc2);
      }
    }
  }

  float* o = out + b * C_ * HW + hw;
  o[0]      = acc0;
  o[HW]     = acc1;
  o[2 * HW] = acc2;
}

extern "C" void kernel_launch(void* const* d_in, const int* in_sizes, int n_in,
                              void* d_out, int out_size, void* d_ws, size_t ws_size,
                              hipStream_t stream) {
  (void)in_sizes; (void)n_in; (void)out_size; (void)d_ws; (void)ws_size;
  const float* img  = (const float*)d_in[0];
  const float* offs = (const float*)d_in[1];
  const float* wts  = (const float*)d_in[2];
  float* out = (float*)d_out;

  const int total_pixels = B_ * HW;          // 1,048,576
  dim3 grid(total_pixels / BLOCK);           // 8192 blocks of 128 (4 wave32)
  adacof_kernel<<<grid, BLOCK, 0, stream>>>(img, offs, wts, out);
}